// SPDRectified_9835475107854
// MI455X (gfx1250) — compile-verified
//
#include <hip/hip_runtime.h>
#include <hip/hip_bf16.h>

// MI455X (gfx1250) block-Jacobi SPD eigensolver + WMMA uTu reconstruction.
//
//  K1 (one WGP / matrix):  A in LDS (256KB) + 8x(32x32) S,Vt scratch (64KB) = 320KB LDS
//     (full WGP LDS). 8 outer sweeps x 15 tournament rounds; each of the 8 waves solves
//     one 32x32 block-pair eigenproblem (scalar Jacobi, cross-lane via __shfl), then
//     applies the 32x32 rotation to A (cols + rows) and to U (d_ws, 128MB < 192MB L2,
//     so panel updates are L2-resident) as 256x32x32 GEMMs on v_wmma_f32_16x16x4_f32.
//     V is kept transposed (Vt[c][k]) so every WMMA frag fetch is a contiguous
//     8B-aligned pair -> ds_load_b64 instead of 2x ds_load_b32.
//     A tile is brought in with GLOBAL_LOAD_ASYNC_TO_LDS_B128 (ASYNCcnt) when the
//     builtin exists. Then: diag -> bitonic sort (desc) -> clamp(eps) -> s.
//  K2: out = diag(s) * (U^T U): 128x128 supertiles; U panels staged TRANSPOSED in LDS
//     with padded stride 258 (8B-aligned b64 frag loads, bank stride 2), K=256 chained
//     f32 WMMA, row-scaled store.

typedef __attribute__((ext_vector_type(2))) float v2f;
typedef __attribute__((ext_vector_type(8))) float v8f;
typedef int v4i __attribute__((vector_size(4 * sizeof(int))));  // matches builtin param

#define WMMA_F32_4(a, b, c) \
  __builtin_amdgcn_wmma_f32_16x16x4_f32(false, (a), false, (b), (short)0, (c), false, false)

#define NMAT 256
#define BATCH 512
#define OUTER_SWEEPS 8
#define INNER_SWEEPS 4
#define PSTRIDE 258  // padded (even) col stride for transposed U panels in K2

#if __has_builtin(__builtin_amdgcn_global_load_async_to_lds_b128)
#define HAVE_ASYNC_LDS 1
#endif

__device__ __forceinline__ void async_copy_b128(const float4* g, float4* l) {
#ifdef HAVE_ASYNC_LDS
  __builtin_amdgcn_global_load_async_to_lds_b128(
      (__attribute__((address_space(1))) v4i*)g,
      (__attribute__((address_space(3))) v4i*)l, 0, 0);
#else
  *l = *g;
#endif
}

__device__ __forceinline__ void async_wait0() {
#ifdef HAVE_ASYNC_LDS
#if __has_builtin(__builtin_amdgcn_s_wait_asynccnt)
  __builtin_amdgcn_s_wait_asynccnt(0);
#else
  asm volatile("s_wait_asynccnt 0" ::: "memory");
#endif
#endif
}

__global__ __launch_bounds__(256) void spd_eig_kernel(const float* __restrict__ A_g,
                                                      const float* __restrict__ epsP,
                                                      float* __restrict__ U_g,
                                                      float* __restrict__ s_g) {
  extern __shared__ float lds[];
  float* sA  = lds;                  // 256*256 floats
  float* sS  = lds + 65536;          // 8 * 32*32
  float* sVt = lds + 65536 + 8192;   // 8 * 32*32 (transposed: Vt[c*32 + r] = V[r][c])

  const int tid  = threadIdx.x;
  const int lane = tid & 31;
  const int wave = tid >> 5;
  const int l15  = lane & 15;
  const int hi   = lane >> 4;  // 0 or 1
  const int b    = blockIdx.x;

  const float* Ain = A_g + (size_t)b * NMAT * NMAT;
  float* U = U_g + (size_t)b * NMAT * NMAT;

  // ---- A -> LDS via async-to-LDS DMA path (B128); U = I in global scratch ----
  {
    const float4* src = (const float4*)Ain;
    float4* dst = (float4*)sA;
    for (int i = tid; i < NMAT * NMAT / 4; i += 256) async_copy_b128(src + i, dst + i);
  }
  for (int i = tid; i < NMAT * NMAT; i += 256) {
    int r = i >> 8, c = i & 255;
    U[i] = (r == c) ? 1.0f : 0.0f;
  }
  async_wait0();
  __syncthreads();

  float* S  = sS + wave * 1024;
  float* Vt = sVt + wave * 1024;

  for (int sweep = 0; sweep < OUTER_SWEEPS; ++sweep) {
    for (int round = 0; round < 15; ++round) {
      // tournament pairing over 16 block indices (fixed = 15)
      int I, J;
      if (wave == 0) { I = 15; J = round % 15; }
      else           { I = (round + wave) % 15; J = (round + 15 - wave) % 15; }

      // ---- extract 32x32 symmetric subproblem, Vt = I ----
      for (int i = lane; i < 1024; i += 32) {
        int r = i >> 5, c = i & 31;
        int gr = (r < 16) ? I * 16 + r : J * 16 + (r - 16);
        int gc = (c < 16) ? I * 16 + c : J * 16 + (c - 16);
        S[i]  = sA[gr * NMAT + gc];
        Vt[i] = (r == c) ? 1.0f : 0.0f;
      }
      __syncthreads();

      // ---- scalar Jacobi on the 32x32 block (tournament over 32, fixed = 31) ----
      for (int isw = 0; isw < INNER_SWEEPS; ++isw) {
        for (int mr = 0; mr < 31; ++mr) {
          int j16 = lane & 15;
          int p, q;
          if (j16 == 0) { p = 31; q = mr % 31; }
          else          { p = (mr + j16) % 31; q = (mr + 31 - j16) % 31; }
          float app = S[p * 32 + p];
          float aqq = S[q * 32 + q];
          float apq = S[p * 32 + q];
          float cr = 1.0f, sr = 0.0f;
          if (fabsf(apq) > 1e-30f) {
            float tau = (aqq - app) / (2.0f * apq);
            float t = (tau >= 0.0f ? 1.0f : -1.0f) / (fabsf(tau) + sqrtf(1.0f + tau * tau));
            cr = rsqrtf(1.0f + t * t);
            sr = t * cr;
          }
          __syncthreads();
          // column rotations on S and Vt (16 pairs x 32 rows, disjoint)
          for (int u = 0; u < 16; ++u) {
            int idx = u * 32 + lane;
            int jj = idx & 15;
            int r  = idx >> 4;
            int pp, qq;
            if (jj == 0) { pp = 31; qq = mr % 31; }
            else         { pp = (mr + jj) % 31; qq = (mr + 31 - jj) % 31; }
            float cj = __shfl(cr, jj, 32);
            float sj = __shfl(sr, jj, 32);
            float x = S[r * 32 + pp], y = S[r * 32 + qq];
            S[r * 32 + pp] = cj * x - sj * y;
            S[r * 32 + qq] = sj * x + cj * y;
            x = Vt[pp * 32 + r]; y = Vt[qq * 32 + r];   // V[r][pp], V[r][qq] transposed
            Vt[pp * 32 + r] = cj * x - sj * y;
            Vt[qq * 32 + r] = sj * x + cj * y;
          }
          __syncthreads();
          // row rotations on S
          for (int u = 0; u < 16; ++u) {
            int idx = u * 32 + lane;
            int jj = idx & 15;
            int cl = idx >> 4;
            int pp, qq;
            if (jj == 0) { pp = 31; qq = mr % 31; }
            else         { pp = (mr + jj) % 31; qq = (mr + 31 - jj) % 31; }
            float cj = __shfl(cr, jj, 32);
            float sj = __shfl(sr, jj, 32);
            float x = S[pp * 32 + cl], y = S[qq * 32 + cl];
            S[pp * 32 + cl] = cj * x - sj * y;
            S[qq * 32 + cl] = sj * x + cj * y;
          }
          __syncthreads();
        }
      }

      // physical column/row of packed index k (0..31); kb even => SCOL(kb+1)=SCOL(kb)+1
#define SCOL(k) (((k) < 16) ? (I * 16 + (k)) : (J * 16 + ((k) - 16)))

      // ---- A[:, IJ] <- A[:, IJ] * V   (256x32x32 GEMM, b64 frag fetches) ----
      for (int m0 = 0; m0 < NMAT; m0 += 16) {
        v8f acc0 = {}; v8f acc1 = {};
        const float* arow = sA + (m0 + l15) * NMAT;
        for (int kk = 0; kk < 8; ++kk) {
          int kb = kk * 4 + hi * 2;
          v2f a  = *(const v2f*)(arow + SCOL(kb));
          v2f b0 = *(const v2f*)(Vt + l15 * 32 + kb);          // V[kb..kb+1][l15]
          v2f b1 = *(const v2f*)(Vt + (16 + l15) * 32 + kb);   // V[kb..kb+1][16+l15]
          acc0 = WMMA_F32_4(a, b0, acc0);
          acc1 = WMMA_F32_4(a, b1, acc1);
        }
        for (int v = 0; v < 8; ++v) {
          int r = m0 + v + 8 * hi;
          sA[r * NMAT + SCOL(l15)]      = acc0[v];
          sA[r * NMAT + SCOL(16 + l15)] = acc1[v];
        }
      }
      __syncthreads();

      // ---- A[IJ, :] <- V^T * A[IJ, :] ----
      for (int n0 = 0; n0 < NMAT; n0 += 16) {
        v8f acc0 = {}; v8f acc1 = {};
        for (int kk = 0; kk < 8; ++kk) {
          int kb = kk * 4 + hi * 2;
          v2f a0 = *(const v2f*)(Vt + l15 * 32 + kb);          // (V^T)(l15, kb..kb+1)
          v2f a1 = *(const v2f*)(Vt + (16 + l15) * 32 + kb);
          v2f bb;
          bb.x = sA[SCOL(kb) * NMAT + n0 + l15];
          bb.y = sA[SCOL(kb + 1) * NMAT + n0 + l15];
          acc0 = WMMA_F32_4(a0, bb, acc0);
          acc1 = WMMA_F32_4(a1, bb, acc1);
        }
        for (int v = 0; v < 8; ++v) {
          int m = v + 8 * hi;
          sA[SCOL(m) * NMAT + n0 + l15]      = acc0[v];
          sA[SCOL(16 + m) * NMAT + n0 + l15] = acc1[v];
        }
      }
      __syncthreads();

      // ---- U[:, IJ] <- U[:, IJ] * V   (global, L2-resident; b64 global frags) ----
      for (int m0 = 0; m0 < NMAT; m0 += 16) {
        v8f acc0 = {}; v8f acc1 = {};
        const float* urow = U + (m0 + l15) * NMAT;
        for (int kk = 0; kk < 8; ++kk) {
          int kb = kk * 4 + hi * 2;
          v2f a  = *(const v2f*)(urow + SCOL(kb));
          v2f b0 = *(const v2f*)(Vt + l15 * 32 + kb);
          v2f b1 = *(const v2f*)(Vt + (16 + l15) * 32 + kb);
          acc0 = WMMA_F32_4(a, b0, acc0);
          acc1 = WMMA_F32_4(a, b1, acc1);
        }
        for (int v = 0; v < 8; ++v) {
          int r = m0 + v + 8 * hi;
          U[r * NMAT + SCOL(l15)]      = acc0[v];
          U[r * NMAT + SCOL(16 + l15)] = acc1[v];
        }
      }
      __syncthreads();
#undef SCOL
    }
  }

  // ---- eigenvalues = diag(A); bitonic sort descending; clamp(eps) ----
  float* sb = sS;  // reuse scratch
  sb[tid] = sA[tid * NMAT + tid];
  __syncthreads();
  for (int k = 2; k <= 256; k <<= 1) {
    for (int j = k >> 1; j > 0; j >>= 1) {
      int ixj = tid ^ j;
      if (ixj > tid) {
        float x = sb[tid], y = sb[ixj];
        bool desc = ((tid & k) == 0);
        if (desc ? (x < y) : (x > y)) { sb[tid] = y; sb[ixj] = x; }
      }
      __syncthreads();
    }
  }
  float eps = epsP[0];
  s_g[b * NMAT + tid] = fmaxf(sb[tid], eps);
}

// out[b] = diag(s) * (U^T U): 128x128 supertile per block, transposed panels in LDS.
__global__ __launch_bounds__(256) void spd_out_kernel(const float* __restrict__ U_g,
                                                      const float* __restrict__ s_g,
                                                      float* __restrict__ out) {
  extern __shared__ float lds[];
  float* P1t = lds;                    // [128][PSTRIDE] : P1t[c][k] = U[k][i0+c]
  float* P2t = lds + 128 * PSTRIDE;    // [128][PSTRIDE] : P2t[c][k] = U[k][j0+c]
  float* sv  = lds + 2 * 128 * PSTRIDE;  // 128 clamped values for rows i0..

  const int tid  = threadIdx.x;
  const int lane = tid & 31;
  const int wave = tid >> 5;
  const int l15  = lane & 15;
  const int hi   = lane >> 4;
  const int b    = blockIdx.x;
  const int st   = blockIdx.y;
  const int i0   = (st >> 1) * 128;
  const int j0   = (st & 1) * 128;

  const float* U = U_g + (size_t)b * NMAT * NMAT;

  // load + transpose panels: float4 global reads, 4 scattered LDS stores each
  for (int idx = tid; idx < 256 * 32; idx += 256) {  // 32 float4 per 128-wide row
    int k  = idx >> 5;
    int c0 = (idx & 31) * 4;
    float4 u1 = *(const float4*)(U + k * NMAT + i0 + c0);
    float4 u2 = *(const float4*)(U + k * NMAT + j0 + c0);
    P1t[(c0 + 0) * PSTRIDE + k] = u1.x;
    P1t[(c0 + 1) * PSTRIDE + k] = u1.y;
    P1t[(c0 + 2) * PSTRIDE + k] = u1.z;
    P1t[(c0 + 3) * PSTRIDE + k] = u1.w;
    P2t[(c0 + 0) * PSTRIDE + k] = u2.x;
    P2t[(c0 + 1) * PSTRIDE + k] = u2.y;
    P2t[(c0 + 2) * PSTRIDE + k] = u2.z;
    P2t[(c0 + 3) * PSTRIDE + k] = u2.w;
  }
  if (tid < 128) sv[tid] = s_g[b * NMAT + i0 + tid];
  __syncthreads();

  for (int t = wave; t < 64; t += 8) {
    int m0 = (t >> 3) * 16;
    int n0 = (t & 7) * 16;
    v8f acc = {};
    for (int kk = 0; kk < 64; ++kk) {
      int kb = kk * 4 + hi * 2;
      v2f a  = *(const v2f*)(P1t + (m0 + l15) * PSTRIDE + kb);
      v2f bb = *(const v2f*)(P2t + (n0 + l15) * PSTRIDE + kb);
      acc = WMMA_F32_4(a, bb, acc);
    }
    for (int v = 0; v < 8; ++v) {
      int ri = m0 + v + 8 * hi;
      out[(size_t)b * NMAT * NMAT + (size_t)(i0 + ri) * NMAT + (j0 + n0 + l15)] =
          sv[ri] * acc[v];
    }
  }
}

extern "C" void kernel_launch(void* const* d_in, const int* in_sizes, int n_in,
                              void* d_out, int out_size, void* d_ws, size_t ws_size,
                              hipStream_t stream) {
  (void)in_sizes; (void)n_in; (void)out_size; (void)ws_size;
  const float* A   = (const float*)d_in[0];
  const float* eps = (const float*)d_in[1];
  float* U = (float*)d_ws;                                                // 128 MB
  float* s = (float*)((char*)d_ws + (size_t)BATCH * NMAT * NMAT * 4);     // 512 KB
  float* out = (float*)d_out;

  const size_t lds1 = (size_t)(65536 + 8192 + 8192) * 4;      // 320 KB (full WGP LDS)
  const size_t lds2 = (size_t)(2 * 128 * PSTRIDE + 128) * 4;  // ~258.5 KB

  (void)hipFuncSetAttribute(reinterpret_cast<const void*>(spd_eig_kernel),
                            hipFuncAttributeMaxDynamicSharedMemorySize, (int)lds1);
  (void)hipFuncSetAttribute(reinterpret_cast<const void*>(spd_out_kernel),
                            hipFuncAttributeMaxDynamicSharedMemorySize, (int)lds2);

  spd_eig_kernel<<<BATCH, 256, lds1, stream>>>(A, eps, U, s);
  spd_out_kernel<<<dim3(BATCH, 4), 256, lds2, stream>>>(U, s, out);
}